// Set2Set_16243566313856
// MI455X (gfx1250) — compile-verified
//
#include <hip/hip_runtime.h>
#include <hip/hip_bf16.h>

// Set2Set (B=1024, N=128, d=128, 3 processing steps), fp32 end-to-end.
// One block = 16 molecules; whole recurrence runs inside the block (no
// inter-block sync needed since molecules are independent).
// LSTM gate GEMM [16 x 384] @ [384 x 512] runs on V_WMMA_F32_16X16X4_F32.

typedef float v2f __attribute__((ext_vector_type(2)));
typedef float v8f __attribute__((ext_vector_type(8)));

#define XS 385   // row stride of x = [q | r | h]  (16 x 384 padded)
#define GS 513   // row stride of gates (16 x 512 padded)
#define AS 129   // row stride of attention weights (16 x 128 padded)

__global__ __launch_bounds__(256) void set2set_kernel(
    const float* __restrict__ y,        // [B,128,128]
    const int*   __restrict__ lengths,  // [B]
    const float* __restrict__ Wih,      // [512,256]
    const float* __restrict__ Whh,      // [512,128]
    const float* __restrict__ bih,      // [512]
    const float* __restrict__ bhh,      // [512]
    const float* __restrict__ Wm,       // [256]
    const float* __restrict__ bm,       // [1]
    const float* __restrict__ mean,     // [1]
    const float* __restrict__ stddev,   // [1]
    float* __restrict__ out)            // [B]
{
    __shared__ float s_x[16 * XS];      // [q(128) | r(128) | h(128)] per molecule
    __shared__ float s_g[16 * GS];      // gates; first 16*AS floats reused as softmax a[]
    __shared__ float s_bias[512];       // b_ih + b_hh
    float* s_a = s_g;                   // alias: a[] live only after gates consumed

    const int tid  = threadIdx.x;
    const int lane = tid & 31;
    const int wave = tid >> 5;
    const int col  = lane & 15;         // M-row for A frag / N-col for B,D frags
    const int kh   = lane >> 4;         // half-wave selector (K pair / M+8)
    const int b0   = blockIdx.x * 16;

    for (int i = tid; i < 16 * XS; i += 256) s_x[i] = 0.f;      // q_star=0, h=0
    for (int i = tid; i < 512; i += 256)     s_bias[i] = bih[i] + bhh[i];
    float creg[8];                                               // c state, fixed (m,j) map
    #pragma unroll
    for (int it = 0; it < 8; ++it) creg[it] = 0.f;
    __syncthreads();

    for (int step = 0; step < 3; ++step) {
        // ---- gates[16,512] = [q_star(256) | h(128)] @ [Wih^T ; Whh^T]  (WMMA f32 16x16x4)
        v8f acc[4] = {};
        for (int k0 = 0; k0 < 256; k0 += 4) {                    // q_star * Wih^T
            const int ka = k0 + 2 * kh;
            v2f a; a[0] = s_x[col * XS + ka]; a[1] = s_x[col * XS + ka + 1];
            #pragma unroll
            for (int t = 0; t < 4; ++t) {
                const int n = (wave * 4 + t) * 16 + col;
                v2f bb; bb[0] = Wih[n * 256 + ka]; bb[1] = Wih[n * 256 + ka + 1];
                acc[t] = __builtin_amdgcn_wmma_f32_16x16x4_f32(
                    false, a, false, bb, (short)0, acc[t], false, false);
            }
        }
        for (int k0 = 256; k0 < 384; k0 += 4) {                  // h * Whh^T
            const int ka = k0 + 2 * kh;
            v2f a; a[0] = s_x[col * XS + ka]; a[1] = s_x[col * XS + ka + 1];
            #pragma unroll
            for (int t = 0; t < 4; ++t) {
                const int n = (wave * 4 + t) * 16 + col;
                v2f bb; bb[0] = Whh[n * 128 + ka - 256]; bb[1] = Whh[n * 128 + ka - 255];
                acc[t] = __builtin_amdgcn_wmma_f32_16x16x4_f32(
                    false, a, false, bb, (short)0, acc[t], false, false);
            }
        }
        #pragma unroll
        for (int t = 0; t < 4; ++t) {
            const int n = (wave * 4 + t) * 16 + col;
            #pragma unroll
            for (int r = 0; r < 8; ++r)
                s_g[(r + 8 * kh) * GS + n] = acc[t][r];          // D: M=r+8*kh, N=n
        }
        __syncthreads();

        // ---- LSTM pointwise (gate order i, f, g, o) ----
        #pragma unroll
        for (int it = 0; it < 8; ++it) {
            const int idx = tid + it * 256;
            const int m = idx >> 7, j = idx & 127;
            const float gi = s_g[m * GS +       j] + s_bias[      j];
            const float gf = s_g[m * GS + 128 + j] + s_bias[128 + j];
            const float gg = s_g[m * GS + 256 + j] + s_bias[256 + j];
            const float go = s_g[m * GS + 384 + j] + s_bias[384 + j];
            const float iv = 1.f / (1.f + expf(-gi));
            const float fv = 1.f / (1.f + expf(-gf));
            const float gv = tanhf(gg);
            const float ov = 1.f / (1.f + expf(-go));
            const float c2 = fv * creg[it] + iv * gv;
            const float h2 = ov * tanhf(c2);
            creg[it] = c2;
            s_x[m * XS +       j] = h2;                          // q = h2
            s_x[m * XS + 256 + j] = h2;                          // h = h2
        }
        __syncthreads();

        // ---- attention: each wave owns 2 molecules ----
        for (int mi = 0; mi < 2; ++mi) {
            const int m = wave * 2 + mi;
            const int b = b0 + m;
            const int len = lengths[b];
            const float* yb = y + (size_t)b * (128 * 128);
            float ev[4];
            float lmax = -3.402823466e38f;
            #pragma unroll
            for (int cc = 0; cc < 4; ++cc) {                     // e[n] = <y[b,n,:], q>
                const int n = cc * 32 + lane;
                const float* yr = yb + n * 128;
                float dot = 0.f;
                for (int d = 0; d < 128; d += 4) {
                    const float4 yv = *(const float4*)(yr + d);
                    dot += yv.x * s_x[m * XS + d]     + yv.y * s_x[m * XS + d + 1]
                         + yv.z * s_x[m * XS + d + 2] + yv.w * s_x[m * XS + d + 3];
                }
                ev[cc] = dot;
                if (n < len) lmax = fmaxf(lmax, dot);
            }
            #pragma unroll
            for (int o = 16; o > 0; o >>= 1) lmax = fmaxf(lmax, __shfl_xor(lmax, o, 32));
            float lsum = 0.f, pv[4];
            #pragma unroll
            for (int cc = 0; cc < 4; ++cc) {
                const int n = cc * 32 + lane;
                const float p = (n < len) ? expf(ev[cc] - lmax) : 0.f;
                pv[cc] = p; lsum += p;
            }
            #pragma unroll
            for (int o = 16; o > 0; o >>= 1) lsum += __shfl_xor(lsum, o, 32);
            const float inv = 1.f / lsum;
            #pragma unroll
            for (int cc = 0; cc < 4; ++cc)
                s_a[m * AS + cc * 32 + lane] = pv[cc] * inv;     // same-wave LDS, no barrier

            // r[d] = sum_n a[n] * y[b,n,d] ; lane owns 4 consecutive d
            const int dbase = lane * 4;
            float r0 = 0.f, r1 = 0.f, r2 = 0.f, r3 = 0.f;
            for (int n = 0; n < 128; ++n) {
                const float av = s_a[m * AS + n];
                const float4 yv = *(const float4*)(yb + n * 128 + dbase);
                r0 += av * yv.x; r1 += av * yv.y; r2 += av * yv.z; r3 += av * yv.w;
            }
            s_x[m * XS + 128 + dbase + 0] = r0;
            s_x[m * XS + 128 + dbase + 1] = r1;
            s_x[m * XS + 128 + dbase + 2] = r2;
            s_x[m * XS + 128 + dbase + 3] = r3;
        }
        __syncthreads();
    }

    // ---- out[b] = (q_star . Wm + bm) * stddev + mean ----
    for (int mi = 0; mi < 2; ++mi) {
        const int m = wave * 2 + mi;
        const int b = b0 + m;
        float p = 0.f;
        for (int j = lane; j < 256; j += 32) p += s_x[m * XS + j] * Wm[j];
        #pragma unroll
        for (int o = 16; o > 0; o >>= 1) p += __shfl_xor(p, o, 32);
        if (lane == 0) out[b] = (p + bm[0]) * stddev[0] + mean[0];
    }
}

extern "C" void kernel_launch(void* const* d_in, const int* in_sizes, int n_in,
                              void* d_out, int out_size, void* d_ws, size_t ws_size,
                              hipStream_t stream) {
    (void)n_in; (void)out_size; (void)d_ws; (void)ws_size;
    const float* y       = (const float*)d_in[0];
    const int*   lengths = (const int*)  d_in[1];
    const float* Wih     = (const float*)d_in[2];
    const float* Whh     = (const float*)d_in[3];
    const float* bih     = (const float*)d_in[4];
    const float* bhh     = (const float*)d_in[5];
    const float* Wm      = (const float*)d_in[6];
    const float* bm      = (const float*)d_in[7];
    const float* mean    = (const float*)d_in[8];
    const float* stddev  = (const float*)d_in[9];
    float* out = (float*)d_out;

    const int B = in_sizes[1];                    // lengths has B elements (1024)
    set2set_kernel<<<dim3(B / 16), dim3(256), 0, stream>>>(
        y, lengths, Wih, Whh, bih, bhh, Wm, bm, mean, stddev, out);
}